// RateRNN_dale_10728828305923
// MI455X (gfx1250) — compile-verified
//
#include <hip/hip_runtime.h>

#define B_ 64
#define S_ 512
#define I_ 128
#define H_ 1024
#define O_ 128
#define ALPHA 0.1f          // DT/TAU = 10/100

typedef float v2f __attribute__((ext_vector_type(2)));
typedef float v8f __attribute__((ext_vector_type(8)));

// D = A(16x4 f32) * B(4x16 f32) + C(16x16 f32), full fp32 matrix core op.
static __device__ __forceinline__ v8f wmma4(v2f a, v2f b, v8f c) {
    return __builtin_amdgcn_wmma_f32_16x16x4_f32(
        /*neg_a=*/false, a, /*neg_b=*/false, b,
        /*c_mod=*/(short)0, c, /*reuse_a=*/false, /*reuse_b=*/false);
}

static __device__ __forceinline__ float sigm(float x) {
    return 1.0f / (1.0f + __expf(-x));
}

// ---------------------------------------------------------------------------
// Kernel 0: h_cur = h0 ; sh0 = sigmoid(h0)
// ---------------------------------------------------------------------------
__global__ __launch_bounds__(256)
void rrnn_init(const float* __restrict__ h0, float* __restrict__ h_cur,
               float* __restrict__ sh0) {
    int i = blockIdx.x * 256 + threadIdx.x;
    if (i < B_ * H_) {
        float v = h0[i];
        h_cur[i] = v;
        sh0[i]   = sigm(v);
    }
}

// ---------------------------------------------------------------------------
// Kernel 1: hidden[b,s,:] (pre-pass) = ALPHA * (x @ W_ih^T) + 0.01 * noise
//   M = B*S = 32768, K = I = 128, N = H = 1024. Wave tile 16x64.
//   grid = (2048, 2), block = 256 (8 waves -> 8 N-columns)
// ---------------------------------------------------------------------------
__global__ __launch_bounds__(256)
void rrnn_pre(const float* __restrict__ x, const float* __restrict__ W_ih,
              const float* __restrict__ noise, float* __restrict__ hidden) {
    const int wave = threadIdx.x >> 5;
    const int lane = threadIdx.x & 31;
    const int hi = lane >> 4, lr = lane & 15;
    const int m_base = blockIdx.x * 16;
    const int n_base = (blockIdx.y * 8 + wave) * 64;

    v8f acc[4] = {};
    const float* arow = x + (m_base + lr) * I_ + 2 * hi;
    #pragma unroll 4
    for (int k = 0; k < I_; k += 4) {
        v2f a = *(const v2f*)(arow + k);
        #pragma unroll
        for (int j = 0; j < 4; ++j) {
            v2f b = *(const v2f*)(W_ih + (n_base + 16 * j + lr) * I_ + 2 * hi + k);
            acc[j] = wmma4(a, b, acc[j]);
        }
    }
    const int row_lo = m_base + 8 * hi;
    #pragma unroll
    for (int j = 0; j < 4; ++j) {
        const int col = n_base + 16 * j + lr;
        #pragma unroll
        for (int r = 0; r < 8; ++r) {
            const int row = row_lo + r;       // row = b*S + s
            const int b = row >> 9;           // /S (512)
            const int s = row & 511;
            float v = ALPHA * acc[j][r]
                    + 0.01f * noise[((s << 6) + b) * H_ + col]; // noise[S,B,1,H]
            hidden[row * H_ + col] = v;
        }
    }
}

// ---------------------------------------------------------------------------
// Kernel 2 (x512, sequential): one recurrence step, split-K for parallelism.
//   rec  = relu(sigmoid(h) @ W_hh^T) * sign
//   h_n  = (1-ALPHA)*h + ALPHA*rec + hidden[b,t,:]   (pre-pass value, in place)
//   also: h_cur = h_n ; sh_write = sigmoid(h_n)  (A operand for step t+1)
//
//   M = B = 64, K = N = H = 1024.  Block = 8 waves; all 8 waves compute the
//   SAME 16x64 output tile over disjoint K-ranges of 128, then combine the
//   partials through LDS with a fixed-order (deterministic) sum.
//   grid = (4 M-tiles, 16 N-columns) -> 64 blocks = 512 waves per step.
// ---------------------------------------------------------------------------
__global__ __launch_bounds__(256)
void rrnn_step(const float* __restrict__ W_hh, const float* __restrict__ sign,
               float* __restrict__ hidden, float* __restrict__ h_cur,
               const float* __restrict__ sh_read, float* __restrict__ sh_write,
               int t) {
    __shared__ float red[8 * 1024];   // red[w][c][lane], c = j*8+r : 32 KB

    const int wave = threadIdx.x >> 5;
    const int lane = threadIdx.x & 31;
    const int hi = lane >> 4, lr = lane & 15;
    const int m_base = blockIdx.x * 16;     // batch rows
    const int n_base = blockIdx.y * 64;     // hidden cols
    const int k_base = wave * 128;          // this wave's K-range

    v8f acc[4] = {};
    const float* arow = sh_read + (m_base + lr) * H_ + 2 * hi + k_base;
    const float* brow = W_hh + 2 * hi + k_base;
    #pragma unroll 4
    for (int k = 0; k < 128; k += 4) {
        v2f a = *(const v2f*)(arow + k);
        #pragma unroll
        for (int j = 0; j < 4; ++j) {
            v2f b = *(const v2f*)(brow + (n_base + 16 * j + lr) * H_ + k);
            acc[j] = wmma4(a, b, acc[j]);
        }
    }
    // Deposit partial tile: lane-contiguous -> conflict-free on 64 banks.
    #pragma unroll
    for (int j = 0; j < 4; ++j)
        #pragma unroll
        for (int r = 0; r < 8; ++r)
            red[wave * 1024 + (j * 8 + r) * 32 + lane] = acc[j][r];
    __syncthreads();

    // Each thread reduces 4 of the 1024 tile elements (fixed order: w=0..7).
    #pragma unroll
    for (int q = 0; q < 4; ++q) {
        const int c = wave + 8 * q;     // 0..31
        const int j = c >> 3;           // 0..3
        const int r = c & 7;            // 0..7
        float v = 0.0f;
        #pragma unroll
        for (int w = 0; w < 8; ++w) v += red[w * 1024 + c * 32 + lane];

        const int b    = m_base + r + 8 * hi;        // batch index
        const int col  = n_base + 16 * j + lr;
        const int hix  = b * H_ + col;
        const int didx = (b * S_ + t) * H_ + col;
        float rec = fmaxf(v, 0.0f) * sign[col];
        float hn  = (1.0f - ALPHA) * h_cur[hix] + ALPHA * rec + hidden[didx];
        hidden[didx]  = hn;
        h_cur[hix]    = hn;
        sh_write[hix] = sigm(hn);
    }
}

// ---------------------------------------------------------------------------
// Kernel 3: out = softmax(sigmoid(hidden) @ W_ho^T + b_ho, axis=O)
//   M = 32768, K = 1024, N = O = 128. Wave owns 16 rows x full N (8 acc
//   tiles); softmax over O done in-wave with shfl_xor (masks < 16 stay inside
//   each 16-lane half, matching the C-layout N = lane&15).
//   grid = 256, block = 256 (8 waves -> 8 M-tiles per block).
// ---------------------------------------------------------------------------
__global__ __launch_bounds__(256)
void rrnn_out(const float* __restrict__ hidden, const float* __restrict__ W_ho,
              const float* __restrict__ b_ho, float* __restrict__ out) {
    const int wave = threadIdx.x >> 5;
    const int lane = threadIdx.x & 31;
    const int hi = lane >> 4, lr = lane & 15;
    const int m_base = (blockIdx.x * 8 + wave) * 16;

    v8f acc[8] = {};
    const float* arow = hidden + (m_base + lr) * H_ + 2 * hi;
    for (int k = 0; k < H_; k += 4) {
        v2f a = *(const v2f*)(arow + k);
        a.x = sigm(a.x);
        a.y = sigm(a.y);
        #pragma unroll
        for (int j = 0; j < 8; ++j) {
            v2f b = *(const v2f*)(W_ho + (16 * j + lr) * H_ + 2 * hi + k);
            acc[j] = wmma4(a, b, acc[j]);
        }
    }
    float bias[8];
    #pragma unroll
    for (int j = 0; j < 8; ++j) bias[j] = b_ho[16 * j + lr];

    const int row_lo = m_base + 8 * hi;
    #pragma unroll
    for (int r = 0; r < 8; ++r) {
        float mx = -3.402823466e38f;
        #pragma unroll
        for (int j = 0; j < 8; ++j) mx = fmaxf(mx, acc[j][r] + bias[j]);
        #pragma unroll
        for (int d = 1; d < 16; d <<= 1) mx = fmaxf(mx, __shfl_xor(mx, d, 32));
        float sum = 0.0f;
        #pragma unroll
        for (int j = 0; j < 8; ++j) {
            float e = __expf(acc[j][r] + bias[j] - mx);
            acc[j][r] = e;
            sum += e;
        }
        #pragma unroll
        for (int d = 1; d < 16; d <<= 1) sum += __shfl_xor(sum, d, 32);
        const float inv = 1.0f / sum;
        const int row = row_lo + r;
        #pragma unroll
        for (int j = 0; j < 8; ++j)
            out[row * O_ + 16 * j + lr] = acc[j][r] * inv;
    }
}

// ---------------------------------------------------------------------------
extern "C" void kernel_launch(void* const* d_in, const int* in_sizes, int n_in,
                              void* d_out, int out_size, void* d_ws, size_t ws_size,
                              hipStream_t stream) {
    const float* x     = (const float*)d_in[0];  // [B,S,I]
    const float* h0    = (const float*)d_in[1];  // [B,1,H]
    const float* W_ih  = (const float*)d_in[2];  // [H,I]
    const float* W_hh  = (const float*)d_in[3];  // [H,H]
    const float* W_ho  = (const float*)d_in[4];  // [O,H]
    const float* b_ho  = (const float*)d_in[5];  // [O]
    const float* sign  = (const float*)d_in[6];  // [H]
    const float* noise = (const float*)d_in[7];  // [S,B,1,H]

    float* outp   = (float*)d_out;                        // [B,S,O]
    float* hidden = outp + (size_t)B_ * S_ * O_;          // [B,S,H]

    float* wsf   = (float*)d_ws;
    float* h_cur = wsf;                 // [B,H]
    float* sh0   = wsf + B_ * H_;       // sigmoid(h) ping
    float* sh1   = wsf + 2 * B_ * H_;   // sigmoid(h) pong

    rrnn_init<<<(B_ * H_ + 255) / 256, 256, 0, stream>>>(h0, h_cur, sh0);
    rrnn_pre<<<dim3((B_ * S_) / 16, 2), 256, 0, stream>>>(x, W_ih, noise, hidden);
    for (int t = 0; t < S_; ++t) {
        const float* shr = (t & 1) ? sh1 : sh0;
        float*       shw = (t & 1) ? sh0 : sh1;
        rrnn_step<<<dim3(4, 16), 256, 0, stream>>>(W_hh, sign, hidden, h_cur,
                                                   shr, shw, t);
    }
    rrnn_out<<<256, 256, 0, stream>>>(hidden, W_ho, b_ho, outp);
}